// NeighborhoodEvolutionBank_76836964926329
// MI455X (gfx1250) — compile-verified
//
#include <hip/hip_runtime.h>
#include <stdint.h>

// Problem constants (fixed by the reference).
#define NUM_NODES 50000
#define WINDOW    8
#define DIM       256
#define BATCH     8192

#define BANK_ELEMS ((size_t)NUM_NODES * WINDOW * DIM)  // 102,400,000 floats
#define TS_ELEMS   ((size_t)NUM_NODES * WINDOW)        // 400,000 floats

// Native clang vector so __builtin_nontemporal_* accepts it (HIP's float4 is
// a class type and is rejected by the builtin's frontend check).
typedef float vfloat4 __attribute__((ext_vector_type(4)));

// ---------------------------------------------------------------------------
// Bulk bank copy (409.6 MB) through the CDNA5 async global<->LDS DMA path,
// with non-temporal hints so the read-once/write-once stream doesn't churn
// the 192 MB L2. Each block stages a 16 KB tile in LDS: per thread 4x async
// B128 loads, s_wait_asynccnt 0, 4x async B128 stores. Each wave only stores
// LDS regions its own lanes loaded, so the per-wave ASYNCcnt wait is the only
// synchronization needed (no __syncthreads).
// ---------------------------------------------------------------------------
__global__ __launch_bounds__(256) void bank_copy_async(const float* __restrict__ src,
                                                       float* __restrict__ dst) {
    __shared__ __align__(16) float stage[256 * 4 * 4]; // 16 KB
    const unsigned t = threadIdx.x;
    const size_t v4base = (size_t)blockIdx.x * 1024;   // tile base, float4 units
    // Low 32 bits of a generic pointer to __shared__ == LDS byte address.
    const uint32_t lds0 = (uint32_t)(uintptr_t)&stage[0];

#pragma unroll
    for (int u = 0; u < 4; ++u) {
        uint32_t loff = lds0 + (uint32_t)(u * 256 + t) * 16u;
        const void* g = (const char*)src + (v4base + (size_t)(u * 256 + t)) * 16;
        asm volatile("global_load_async_to_lds_b128 %0, %1, off th:TH_LOAD_NT"
                     :: "v"(loff), "v"(g) : "memory");
    }
    asm volatile("s_wait_asynccnt 0" ::: "memory");
#pragma unroll
    for (int u = 0; u < 4; ++u) {
        uint32_t loff = lds0 + (uint32_t)(u * 256 + t) * 16u;
        void* g = (char*)dst + (v4base + (size_t)(u * 256 + t)) * 16;
        asm volatile("global_store_async_from_lds_b128 %0, %1, off th:TH_STORE_NT"
                     :: "v"(g), "v"(loff) : "memory");
    }
    asm volatile("s_wait_asynccnt 0" ::: "memory");
}

// ---------------------------------------------------------------------------
// Timestamp passthrough copy (1.6 MB): non-temporal B128 streaming copy.
// ---------------------------------------------------------------------------
__global__ __launch_bounds__(256) void ts_copy(const vfloat4* __restrict__ src,
                                               vfloat4* __restrict__ dst, int n4) {
    int i = blockIdx.x * 256 + threadIdx.x;
    if (i < n4) {
        vfloat4 v = __builtin_nontemporal_load(&src[i]);
        __builtin_nontemporal_store(v, &dst[i]);
    }
}

// ---------------------------------------------------------------------------
// Rank + scatter: one wave32 per batch element bi.
//   rank  = #{j < bi : idx[j] == idx[bi]}
//   count = #{j      : idx[j] == idx[bi]}
//   slot  = (ptr[idx[bi]] + rank) & 7
// Only the last WINDOW occurrences per node land (torch sequential-loop
// semantics): winner iff rank + WINDOW >= count; winner ranks span 8
// consecutive values so their slots are distinct mod 8. Winners write a
// 256-float row (2 float4 per lane) + timestamp. The rank==0 occurrence
// records the per-node occurrence count for ptr_new.
// ---------------------------------------------------------------------------
__global__ __launch_bounds__(256) void rank_scatter(const float* __restrict__ nrep,
                                                    const float* __restrict__ tvals,
                                                    const int* __restrict__ ptr,
                                                    const int* __restrict__ idx,
                                                    float* __restrict__ out_bank,
                                                    float* __restrict__ out_ts,
                                                    int* __restrict__ ws_count) {
    __shared__ int sidx[BATCH]; // 32 KB
    for (int i = threadIdx.x; i < BATCH; i += 256) sidx[i] = idx[i];
    __syncthreads();

    const int wave = threadIdx.x >> 5;
    const int lane = threadIdx.x & 31;
    const int bi   = blockIdx.x * 8 + wave; // gridDim.x == BATCH/8
    const int my   = sidx[bi];

    int rank = 0, count = 0;
    for (int j = lane; j < BATCH; j += 32) {
        int eq = (sidx[j] == my);
        count += eq;
        rank  += eq & (j < bi);
    }
    // wave32 butterfly reduction: every lane ends with the full sums.
#pragma unroll
    for (int off = 16; off > 0; off >>= 1) {
        rank  += __shfl_xor(rank,  off, 32);
        count += __shfl_xor(count, off, 32);
    }

    if (rank == 0 && lane == 0) ws_count[my] = count; // unique writer per node

    if (rank + WINDOW >= count) { // "last write wins" winner
        const int slot = (ptr[my] + rank) & (WINDOW - 1); // ptr >= 0, WINDOW=8
        vfloat4* dst = (vfloat4*)(out_bank + ((size_t)my * WINDOW + slot) * DIM);
        const vfloat4* src = (const vfloat4*)(nrep + (size_t)bi * DIM);
        dst[lane]      = src[lane];        // 64 float4 total, 2 per lane
        dst[lane + 32] = src[lane + 32];
        if (lane == 0) out_ts[my * WINDOW + slot] = tvals[bi];
    }
}

// ---------------------------------------------------------------------------
// ptr_new[i] = ptr[i] + occurrences(i)   (emitted as float into d_out tail)
// ---------------------------------------------------------------------------
__global__ __launch_bounds__(256) void finalize_ptr(const int* __restrict__ ptr,
                                                    const int* __restrict__ ws_count,
                                                    float* __restrict__ out_ptr) {
    int i = blockIdx.x * 256 + threadIdx.x;
    if (i < NUM_NODES) out_ptr[i] = (float)(ptr[i] + ws_count[i]);
}

extern "C" void kernel_launch(void* const* d_in, const int* in_sizes, int n_in,
                              void* d_out, int out_size, void* d_ws, size_t ws_size,
                              hipStream_t stream) {
    (void)in_sizes; (void)n_in; (void)out_size; (void)ws_size;
    const float* bank = (const float*)d_in[0];
    const float* ts   = (const float*)d_in[1];
    const float* nrep = (const float*)d_in[2];
    const float* t    = (const float*)d_in[3];
    const int*   ptr  = (const int*)d_in[4];
    const int*   idx  = (const int*)d_in[5];

    float* out_bank = (float*)d_out;
    float* out_ts   = out_bank + BANK_ELEMS;
    float* out_ptr  = out_ts + TS_ELEMS;
    int*   ws_count = (int*)d_ws; // 200 KB of scratch

    // Per-node occurrence counters must start at zero every call.
    (void)hipMemsetAsync(ws_count, 0, NUM_NODES * sizeof(int), stream);

    // 25,600,000 float4 / 1024 per block = exactly 25,000 blocks.
    bank_copy_async<<<25000, 256, 0, stream>>>(bank, out_bank);

    ts_copy<<<((int)(TS_ELEMS / 4) + 255) / 256, 256, 0, stream>>>(
        (const vfloat4*)ts, (vfloat4*)out_ts, (int)(TS_ELEMS / 4));

    // One wave32 per batch element: 8192 / 8 waves-per-block = 1024 blocks.
    rank_scatter<<<BATCH / 8, 256, 0, stream>>>(nrep, t, ptr, idx,
                                                out_bank, out_ts, ws_count);

    finalize_ptr<<<(NUM_NODES + 255) / 256, 256, 0, stream>>>(ptr, ws_count, out_ptr);
}